// SearchDepthGNN_10917806867267
// MI455X (gfx1250) — compile-verified
//
#include <hip/hip_runtime.h>
#include <hip/hip_bf16.h>
#include <math.h>

#define NN 50000
#define NE 800000
#define NG 128
#define DIM 128
#define NLAYERS 4
#define NCLS 10
#define GN_EPS 1e-5f
#define SEG_NPB 8      // nodes per block for batched segment ops (NN % 8 == 0)
#define SCAN_B 256

typedef __attribute__((ext_vector_type(2))) float v2f;
typedef __attribute__((ext_vector_type(8))) float v8f;

// ---------------- utility kernels ----------------

__global__ void zero_kernel(float* __restrict__ p, int n) {
  int i = blockIdx.x * blockDim.x + threadIdx.x;
  if (i < n) p[i] = 0.0f;
}

__global__ void zero_int_kernel(int* __restrict__ p, int n) {
  int i = blockIdx.x * blockDim.x + threadIdx.x;
  if (i < n) p[i] = 0;
}

__global__ void counts_kernel(const int* __restrict__ batch, float* __restrict__ counts) {
  int i = blockIdx.x * blockDim.x + threadIdx.x;
  if (i < NN) atomicAdd(&counts[batch[i]], 1.0f);
}

// ---------------- CSR build (once per call; edges reused across all 4 layers) ----------------

__global__ void deg_kernel(const int* __restrict__ dst, int* __restrict__ deg) {
  int e = blockIdx.x * blockDim.x + threadIdx.x;
  if (e < NE) atomicAdd(&deg[dst[e]], 1);
}

__global__ void scan_block_kernel(const int* __restrict__ deg, int* __restrict__ incl,
                                  int* __restrict__ bsum) {
  __shared__ int sh[SCAN_B];
  int i = blockIdx.x * SCAN_B + threadIdx.x;
  int v = (i < NN) ? deg[i] : 0;
  sh[threadIdx.x] = v;
  __syncthreads();
  for (int off = 1; off < SCAN_B; off <<= 1) {
    int t = (threadIdx.x >= (unsigned)off) ? sh[threadIdx.x - off] : 0;
    __syncthreads();
    sh[threadIdx.x] += t;
    __syncthreads();
  }
  if (i < NN) incl[i] = sh[threadIdx.x];
  if (threadIdx.x == SCAN_B - 1) bsum[blockIdx.x] = sh[SCAN_B - 1];
}

__global__ void scan_bsum_kernel(int* __restrict__ bsum, int nb) {
  __shared__ int sh[SCAN_B];
  int v = (threadIdx.x < (unsigned)nb) ? bsum[threadIdx.x] : 0;
  sh[threadIdx.x] = v;
  __syncthreads();
  for (int off = 1; off < SCAN_B; off <<= 1) {
    int t = (threadIdx.x >= (unsigned)off) ? sh[threadIdx.x - off] : 0;
    __syncthreads();
    sh[threadIdx.x] += t;
    __syncthreads();
  }
  if (threadIdx.x < (unsigned)nb) bsum[threadIdx.x] = sh[threadIdx.x] - v;  // exclusive
}

__global__ void scan_finalize_kernel(const int* __restrict__ deg, const int* __restrict__ incl,
                                     const int* __restrict__ bsum, int* __restrict__ offs,
                                     int* __restrict__ cursor) {
  int i = blockIdx.x * blockDim.x + threadIdx.x;
  if (i < NN) {
    int o = incl[i] - deg[i] + bsum[i >> 8];
    offs[i] = o;
    cursor[i] = o;
  }
  if (i == 0) offs[NN] = NE;
}

__global__ void csr_fill_kernel(const int* __restrict__ src, const int* __restrict__ dst,
                                int* __restrict__ cursor, int* __restrict__ csr_src) {
  int e = blockIdx.x * blockDim.x + threadIdx.x;
  if (e < NE) {
    int pos = atomicAdd(&cursor[dst[e]], 1);
    csr_src[pos] = src[e];
  }
}

// agg[n] = sum over incoming edges of h[src]; one wave per node, 4-way unrolled for MLP
__global__ __launch_bounds__(256) void gather_kernel(const float* __restrict__ h,
                                                     const int* __restrict__ offs,
                                                     const int* __restrict__ csr_src,
                                                     float* __restrict__ agg) {
  int n = blockIdx.x * 8 + (threadIdx.x >> 5);   // NN % 8 == 0, always valid
  int lane = threadIdx.x & 31;
  int e0 = offs[n], e1 = offs[n + 1];
  float4 acc = {0.f, 0.f, 0.f, 0.f};
  int e = e0;
  for (; e + 4 <= e1; e += 4) {
    int s0 = csr_src[e + 0], s1 = csr_src[e + 1];
    int s2 = csr_src[e + 2], s3 = csr_src[e + 3];
    float4 v0 = ((const float4*)(h + (size_t)s0 * DIM))[lane];
    float4 v1 = ((const float4*)(h + (size_t)s1 * DIM))[lane];
    float4 v2 = ((const float4*)(h + (size_t)s2 * DIM))[lane];
    float4 v3 = ((const float4*)(h + (size_t)s3 * DIM))[lane];
    acc.x += v0.x + v1.x + v2.x + v3.x;
    acc.y += v0.y + v1.y + v2.y + v3.y;
    acc.z += v0.z + v1.z + v2.z + v3.z;
    acc.w += v0.w + v1.w + v2.w + v3.w;
  }
  for (; e < e1; ++e) {
    int s = csr_src[e];
    float4 v = ((const float4*)(h + (size_t)s * DIM))[lane];
    acc.x += v.x; acc.y += v.y; acc.z += v.z; acc.w += v.w;
  }
  ((float4*)(agg + (size_t)n * DIM))[lane] = acc;
}

// ---------------- WMMA fp32 GEMM: C = relu((A [+ A2]) @ W + bias) ----------------
// A: [M x 128] row-major, W: [128 x 128] row-major, bias: [128]
// Block = 256 threads = 8 waves; block computes 32 rows x 128 cols.
// Each wave holds two accumulators (row tiles 0-15 and 16-31) sharing its B registers,
// doubling wmma work per B-load. A rows are clamped on load; stores row-guarded.
__global__ __launch_bounds__(256) void gemm_relu_kernel(const float* __restrict__ A,
                                                        const float* __restrict__ A2,
                                                        const float* __restrict__ W,
                                                        const float* __restrict__ bias,
                                                        float* __restrict__ C,
                                                        int M, int fuse_add) {
  __shared__ float a_tile[32][132];  // padded rows: 16B-aligned, conflict-free column reads
  const int rowbase = blockIdx.x * 32;
  const int tid = threadIdx.x;

  #pragma unroll
  for (int it = 0; it < 4; ++it) {
    int f = tid + it * 256;          // float4 slot 0..1023
    int r = f >> 5;                  // row 0..31
    int c4 = (f & 31) << 2;
    int row = rowbase + r;
    int rowc = row < M ? row : M - 1;   // clamp: never read OOB of inputs
    float4 v = *(const float4*)(A + (size_t)rowc * DIM + c4);
    if (fuse_add) {
      float4 u = *(const float4*)(A2 + (size_t)rowc * DIM + c4);
      v.x += u.x; v.y += u.y; v.z += u.z; v.w += u.w;
    }
    *(float4*)&a_tile[r][c4] = v;
  }
  __syncthreads();

  const int wave = tid >> 5;
  const int lane = tid & 31;
  const int grp  = lane >> 4;
  const int lm   = lane & 15;
  const int colbase = wave * 16;

  v8f acc0 = {}, acc1 = {};
  for (int k = 0; k < DIM; k += 4) {
    v2f a0, a1, b;
    b.x  = W[(size_t)(k + 2 * grp + 0) * DIM + colbase + lm];
    b.y  = W[(size_t)(k + 2 * grp + 1) * DIM + colbase + lm];
    a0.x = a_tile[lm][k + 2 * grp + 0];
    a0.y = a_tile[lm][k + 2 * grp + 1];
    a1.x = a_tile[16 + lm][k + 2 * grp + 0];
    a1.y = a_tile[16 + lm][k + 2 * grp + 1];
    acc0 = __builtin_amdgcn_wmma_f32_16x16x4_f32(false, a0, false, b, (short)0, acc0,
                                                 false, false);
    acc1 = __builtin_amdgcn_wmma_f32_16x16x4_f32(false, a1, false, b, (short)0, acc1,
                                                 false, false);
  }

  const int col = colbase + lm;
  const float bv = bias[col];
  #pragma unroll
  for (int r = 0; r < 8; ++r) {
    int row0 = rowbase + grp * 8 + r;      // C/D layout: VGPR r = row r / 8+r per lane half
    int row1 = row0 + 16;
    float v0 = acc0[r] + bv;
    float v1 = acc1[r] + bv;
    if (row0 < M) C[(size_t)row0 * DIM + col] = v0 > 0.0f ? v0 : 0.0f;
    if (row1 < M) C[(size_t)row1 * DIM + col] = v1 > 0.0f ? v1 : 0.0f;
  }
}

// ---------------- GraphNorm pieces (batch sorted: 8 nodes/block, flush on graph change) ----

__global__ void seg_acc8_kernel(const float* __restrict__ h, const int* __restrict__ batch,
                                float* __restrict__ out) {
  int n0 = blockIdx.x * SEG_NPB;
  int d = threadIdx.x;
  float acc = 0.0f;
  int gprev = batch[n0];
  for (int j = 0; j < SEG_NPB; ++j) {
    int n = n0 + j;
    int g = batch[n];               // block-uniform
    if (g != gprev) {
      atomicAdd(&out[gprev * DIM + d], acc);
      acc = 0.0f;
      gprev = g;
    }
    acc += h[(size_t)n * DIM + d];
  }
  atomicAdd(&out[gprev * DIM + d], acc);
}

// o = h - (meansum[g]/count[g])*scale ; out = o ; varsum[g] += o*o   (division fused)
__global__ void center_var8_kernel(const float* __restrict__ h, const int* __restrict__ batch,
                                   const float* __restrict__ meansum,
                                   const float* __restrict__ counts,
                                   const float* __restrict__ scale,
                                   float* __restrict__ out, float* __restrict__ var) {
  int n0 = blockIdx.x * SEG_NPB;
  int d = threadIdx.x;
  float sc = scale[d];
  float acc = 0.0f;
  int gprev = batch[n0];
  float cnt = counts[gprev]; cnt = cnt < 1.0f ? 1.0f : cnt;
  float ms = meansum[gprev * DIM + d] / cnt * sc;
  for (int j = 0; j < SEG_NPB; ++j) {
    int n = n0 + j;
    int g = batch[n];
    if (g != gprev) {
      atomicAdd(&var[gprev * DIM + d], acc);
      acc = 0.0f;
      gprev = g;
      cnt = counts[g]; cnt = cnt < 1.0f ? 1.0f : cnt;
      ms = meansum[g * DIM + d] / cnt * sc;
    }
    float o = h[(size_t)n * DIM + d] - ms;
    out[(size_t)n * DIM + d] = o;
    acc += o * o;
  }
  atomicAdd(&var[gprev * DIM + d], acc);
}

// h = relu(weight*out*rsqrt(varsum[g]/count[g]+eps) + bias)   (division fused)
__global__ void gn_final8_kernel(const float* __restrict__ out, const int* __restrict__ batch,
                                 const float* __restrict__ varsum,
                                 const float* __restrict__ counts,
                                 const float* __restrict__ weight,
                                 const float* __restrict__ bias, float* __restrict__ h) {
  int n0 = blockIdx.x * SEG_NPB;
  int d = threadIdx.x;
  float w = weight[d], b = bias[d];
  int gprev = batch[n0];
  float cnt = counts[gprev]; cnt = cnt < 1.0f ? 1.0f : cnt;
  float rs = rsqrtf(varsum[gprev * DIM + d] / cnt + GN_EPS);
  for (int j = 0; j < SEG_NPB; ++j) {
    int n = n0 + j;
    int g = batch[n];
    if (g != gprev) {
      gprev = g;
      cnt = counts[g]; cnt = cnt < 1.0f ? 1.0f : cnt;
      rs = rsqrtf(varsum[g * DIM + d] / cnt + GN_EPS);
    }
    float v = w * out[(size_t)n * DIM + d] * rs + b;
    h[(size_t)n * DIM + d] = v > 0.0f ? v : 0.0f;
  }
}

// ---------------- classifier head: [NG x DIM] @ [DIM x NCLS] + log_softmax ----------------
__global__ void head_kernel(const float* __restrict__ g2, const float* __restrict__ fw3,
                            const float* __restrict__ fb3, float* __restrict__ out) {
  int g = blockIdx.x;
  int lane = threadIdx.x;  // 32 threads, one wave32
  float part[NCLS];
  #pragma unroll
  for (int c = 0; c < NCLS; ++c) part[c] = 0.0f;
  for (int d = lane; d < DIM; d += 32) {
    float v = g2[g * DIM + d];
    #pragma unroll
    for (int c = 0; c < NCLS; ++c) part[c] += v * fw3[d * NCLS + c];
  }
  #pragma unroll
  for (int c = 0; c < NCLS; ++c)
    for (int off = 16; off > 0; off >>= 1)
      part[c] += __shfl_down(part[c], off, 32);
  if (lane == 0) {
    float logits[NCLS];
    float m = -1e30f;
    #pragma unroll
    for (int c = 0; c < NCLS; ++c) { logits[c] = part[c] + fb3[c]; m = fmaxf(m, logits[c]); }
    float s = 0.0f;
    #pragma unroll
    for (int c = 0; c < NCLS; ++c) s += expf(logits[c] - m);
    float lse = m + logf(s);
    #pragma unroll
    for (int c = 0; c < NCLS; ++c) out[g * NCLS + c] = logits[c] - lse;
  }
}

// ---------------- host-side orchestration ----------------

static inline void zero_launch(float* p, int n, hipStream_t s) {
  zero_kernel<<<(n + 255) / 256, 256, 0, s>>>(p, n);
}

extern "C" void kernel_launch(void* const* d_in, const int* in_sizes, int n_in,
                              void* d_out, int out_size, void* d_ws, size_t ws_size,
                              hipStream_t stream) {
  const float* x       = (const float*)d_in[0];
  const float* gin_w1  = (const float*)d_in[1];
  const float* gin_b1  = (const float*)d_in[2];
  const float* gin_w2  = (const float*)d_in[3];
  const float* gin_b2  = (const float*)d_in[4];
  const float* gn_w    = (const float*)d_in[5];
  const float* gn_b    = (const float*)d_in[6];
  const float* gn_s    = (const float*)d_in[7];
  const float* fw1     = (const float*)d_in[8];
  const float* fb1     = (const float*)d_in[9];
  const float* fw2     = (const float*)d_in[10];
  const float* fb2     = (const float*)d_in[11];
  const float* fw3     = (const float*)d_in[12];
  const float* fb3     = (const float*)d_in[13];
  const int*   eidx    = (const int*)d_in[14];
  const int*   batch   = (const int*)d_in[15];
  const int*   src     = eidx;
  const int*   dst     = eidx + NE;

  const size_t ND = (size_t)NN * DIM;
  char* ws = (char*)d_ws;
  float* buf0    = (float*)ws; ws += ND * sizeof(float);
  float* buf1    = (float*)ws; ws += ND * sizeof(float);
  float* buf2    = (float*)ws; ws += ND * sizeof(float);
  float* meanbf  = (float*)ws; ws += (size_t)NG * DIM * sizeof(float);
  float* varbf   = (float*)ws; ws += (size_t)NG * DIM * sizeof(float);
  float* gbuf    = (float*)ws; ws += (size_t)NG * DIM * sizeof(float);
  float* gtmp1   = (float*)ws; ws += (size_t)NG * DIM * sizeof(float);
  float* gtmp2   = (float*)ws; ws += (size_t)NG * DIM * sizeof(float);
  float* counts  = (float*)ws; ws += (size_t)NG * sizeof(float);
  int*   deg     = (int*)ws;   ws += (size_t)NN * sizeof(int);
  int*   incl    = (int*)ws;   ws += (size_t)NN * sizeof(int);
  int*   bsum    = (int*)ws;   ws += (size_t)SCAN_B * sizeof(int);
  int*   offs    = (int*)ws;   ws += (size_t)(NN + 1) * sizeof(int);
  int*   cursor  = (int*)ws;   ws += (size_t)NN * sizeof(int);
  int*   csr_src = (int*)ws;   ws += (size_t)NE * sizeof(int);

  const int GD   = NG * DIM;
  const int NB   = (NN + SCAN_B - 1) / SCAN_B;   // 196 scan blocks (<= 256)
  const int GEMM_GRID_N = (NN + 31) / 32;        // 1563 (last block row-guarded)
  const int GEMM_GRID_G = (NG + 31) / 32;        // 4

  // ---- CSR by destination (built once; edges identical across layers) ----
  zero_int_kernel<<<(NN + 255) / 256, 256, 0, stream>>>(deg, NN);
  deg_kernel<<<(NE + 255) / 256, 256, 0, stream>>>(dst, deg);
  scan_block_kernel<<<NB, SCAN_B, 0, stream>>>(deg, incl, bsum);
  scan_bsum_kernel<<<1, SCAN_B, 0, stream>>>(bsum, NB);
  scan_finalize_kernel<<<(NN + 255) / 256, 256, 0, stream>>>(deg, incl, bsum, offs, cursor);
  csr_fill_kernel<<<(NE + 255) / 256, 256, 0, stream>>>(src, dst, cursor, csr_src);

  // ---- per-graph node counts ----
  zero_launch(counts, NG, stream);
  counts_kernel<<<(NN + 255) / 256, 256, 0, stream>>>(batch, counts);

  const float* h = x;
  for (int l = 0; l < NLAYERS; ++l) {
    // agg = segment_sum(h[src], dst)  — gather form, no atomics, no zero-fill
    gather_kernel<<<NN / 8, 256, 0, stream>>>(h, offs, csr_src, buf1);
    // h2 = relu((h + agg) @ W1 + b1)
    gemm_relu_kernel<<<GEMM_GRID_N, 256, 0, stream>>>(h, buf1,
                                                      gin_w1 + (size_t)l * DIM * DIM,
                                                      gin_b1 + l * DIM, buf2, NN, 1);
    // h3 = relu(h2 @ W2 + b2)
    gemm_relu_kernel<<<GEMM_GRID_N, 256, 0, stream>>>(buf2, nullptr,
                                                      gin_w2 + (size_t)l * DIM * DIM,
                                                      gin_b2 + l * DIM, buf1, NN, 0);
    // GraphNorm (divisions by counts fused into consumers)
    zero_launch(meanbf, GD, stream);
    seg_acc8_kernel<<<NN / SEG_NPB, DIM, 0, stream>>>(buf1, batch, meanbf);
    zero_launch(varbf, GD, stream);
    center_var8_kernel<<<NN / SEG_NPB, DIM, 0, stream>>>(buf1, batch, meanbf, counts,
                                                         gn_s + l * DIM, buf2, varbf);
    gn_final8_kernel<<<NN / SEG_NPB, DIM, 0, stream>>>(buf2, batch, varbf, counts,
                                                       gn_w + l * DIM, gn_b + l * DIM, buf0);
    h = buf0;
  }

  // ---- sum readout per graph ----
  zero_launch(gbuf, GD, stream);
  seg_acc8_kernel<<<NN / SEG_NPB, DIM, 0, stream>>>(h, batch, gbuf);

  // ---- final MLP ----
  gemm_relu_kernel<<<GEMM_GRID_G, 256, 0, stream>>>(gbuf, nullptr, fw1, fb1, gtmp1, NG, 0);
  gemm_relu_kernel<<<GEMM_GRID_G, 256, 0, stream>>>(gtmp1, nullptr, fw2, fb2, gtmp2, NG, 0);
  head_kernel<<<NG, 32, 0, stream>>>(gtmp2, fw3, fb3, (float*)d_out);
}